// RnCLoss_29429115912409
// MI455X (gfx1250) — compile-verified
//
#include <hip/hip_runtime.h>
#include <hip/hip_bf16.h>
#include <math.h>

// RnC loss for MI455X (gfx1250, wave32).
// n = 4096 (= 2*2048), d = 256, L = 1, t = 2.0
//
// Pipeline:
//   K0: r[i] = ||f_i||^2, llab[i] = labels[i mod 2048]
//   K1: f32 WMMA GEMM (V_WMMA_F32_16X16X4_F32) -> logits = -sqrt(max(r_i+r_j-2G,0))/t
//       (row-max shift skipped: it cancels analytically in lo - log(denom))
//   K2: per row: bitonic sort by label-diff, suffix scan of exp(lo),
//       lower_bound per element (exact '>=' tie semantics), row partial sum
//   K3: deterministic final reduction -> d_out[0]
//
// Workspace layout (needs ~64.05 MB):
//   [0, 64MB)      lo matrix (n*n f32)
//   then r (n f32), llab (n f32), partial (n f32)

#define N_TOT  4096
#define HALF_N 2048
#define DIM    256
#define TEMP   2.0f

typedef __attribute__((ext_vector_type(2))) float v2f;
typedef __attribute__((ext_vector_type(8))) float v8f;

__device__ __forceinline__ const float* feat_row(const float* z1, const float* z2, int i) {
    return (i < HALF_N) ? (z1 + (size_t)i * DIM) : (z2 + (size_t)(i - HALF_N) * DIM);
}

// ---------------- K0: row norms + tiled labels ----------------
__global__ void rnc_prep(const float* __restrict__ z1, const float* __restrict__ z2,
                         const float* __restrict__ labels,
                         float* __restrict__ r, float* __restrict__ llab) {
    int i = blockIdx.x * blockDim.x + threadIdx.x;
    if (i >= N_TOT) return;
    const float* row = feat_row(z1, z2, i);
    float acc = 0.0f;
    #pragma unroll 8
    for (int k = 0; k < DIM; ++k) {
        float v = row[k];
        acc = fmaf(v, v, acc);
    }
    r[i] = acc;
    llab[i] = labels[(i < HALF_N) ? i : (i - HALF_N)];
}

// ---------------- K1: f32 WMMA GEMM -> logits ----------------
// One wave per 16x16 tile; 64 x V_WMMA_F32_16X16X4_F32 over K=256.
// A 16x4 f32 frag: lanes 0-15 -> M=lane, K={k0,k0+1}; lanes 16-31 -> K={k0+2,k0+3}.
// B 4x16 frag mirrors A with columns (G = F F^T, so B rows come from feats too).
__global__ void rnc_gemm_logits(const float* __restrict__ z1, const float* __restrict__ z2,
                                const float* __restrict__ r,
                                float* __restrict__ lo) {
    const int lane = threadIdx.x & 31;
    const int wave = threadIdx.x >> 5;
    const int tile = blockIdx.x * (blockDim.x >> 5) + wave;   // 0..65535
    const int tm = tile >> 8;          // 256 tiles per dim
    const int tn = tile & 255;
    const int i0 = tm * 16;
    const int j0 = tn * 16;

    const int mn   = lane & 15;
    const int koff = (lane >> 4) << 1;     // 0 or 2

    const float* arow = feat_row(z1, z2, i0 + mn);
    const float* brow = feat_row(z1, z2, j0 + mn);

    v8f acc = {0.f, 0.f, 0.f, 0.f, 0.f, 0.f, 0.f, 0.f};
    #pragma unroll 8
    for (int k0 = 0; k0 < DIM; k0 += 4) {
        v2f a = *(const v2f*)(arow + k0 + koff);
        v2f b = *(const v2f*)(brow + k0 + koff);
        acc = __builtin_amdgcn_wmma_f32_16x16x4_f32(
            /*neg_a=*/false, a, /*neg_b=*/false, b,
            /*c_mod=*/(short)0, acc, /*reuse_a=*/false, /*reuse_b=*/false);
    }

    // C/D layout: lane = n (0-15 twice); VGPR v -> row (lane>=16 ? 8 : 0) + v
    const int n = mn;
    const int mbase = (lane >> 4) * 8;
    const int j = j0 + n;
    const float rj = r[j];
    #pragma unroll
    for (int v = 0; v < 8; ++v) {
        int i = i0 + mbase + v;
        float sq = fmaxf(r[i] + rj - 2.0f * acc[v], 0.0f);
        float l = (i == j) ? (-1.0f / TEMP) : (-sqrtf(sq) / TEMP);  // diag value unused
        lo[(size_t)i * N_TOT + j] = l;
    }
}

// ---------------- K2: per-row sort + suffix scan + lower_bound ----------------
__global__ void rnc_row(const float* __restrict__ lo, const float* __restrict__ llab,
                        float* __restrict__ partial) {
    __shared__ float skey[N_TOT];   // 16 KB: |lab_i - lab_j|
    __shared__ float sval[N_TOT];   // 16 KB: exp(lo) -> suffix sums
    __shared__ float red[256];

    const int i   = blockIdx.x;
    const int tid = threadIdx.x;
    const float li = llab[i];
    const float* lorow = lo + (size_t)i * N_TOT;

    // Phase 1: keys + exponentials (diag ex=0 removes it from every denominator)
    float lo_sum = 0.0f;
    for (int p = tid; p < N_TOT; p += 256) {
        float l = lorow[p];
        skey[p] = fabsf(llab[p] - li);
        float e = 0.0f;
        if (p != i) {
            e = expf(l);
            lo_sum += l;                 // sum of lo over the 4095 real anchors
        }
        sval[p] = e;
    }
    __syncthreads();

    // Phase 2: bitonic sort ascending by key (payload = exp value)
    for (int k = 2; k <= N_TOT; k <<= 1) {
        for (int j = k >> 1; j > 0; j >>= 1) {
            for (int idx = tid; idx < N_TOT; idx += 256) {
                int partner = idx ^ j;
                if (partner > idx) {
                    bool up = ((idx & k) == 0);
                    float ka = skey[idx], kb = skey[partner];
                    bool sw = up ? (ka > kb) : (ka < kb);
                    if (sw) {
                        skey[idx] = kb; skey[partner] = ka;
                        float va = sval[idx], vb = sval[partner];
                        sval[idx] = vb;  sval[partner] = va;
                    }
                }
            }
            __syncthreads();
        }
    }

    // Phase 3: in-place suffix scan (right-to-left Hillis-Steele) of sval
    for (int d = 1; d < N_TOT; d <<= 1) {
        float tmp[16];
        #pragma unroll
        for (int t = 0; t < 16; ++t) {
            int q = tid + t * 256 + d;
            tmp[t] = (q < N_TOT) ? sval[q] : 0.0f;
        }
        __syncthreads();
        #pragma unroll
        for (int t = 0; t < 16; ++t) sval[tid + t * 256] += tmp[t];
        __syncthreads();
    }

    // Phase 4: denom_p = suffix[lower_bound(key_p)]  (== searchsorted 'left')
    float nld = 0.0f;
    for (int p = tid; p < N_TOT; p += 256) {
        float v = skey[p];
        int a = 0, b = N_TOT;
        while (a < b) {
            int mid = (a + b) >> 1;
            if (skey[mid] < v) a = mid + 1; else b = mid;
        }
        nld -= logf(sval[a]);
    }

    // Deterministic row reduction; +log(total) cancels the spurious diagonal term
    red[tid] = lo_sum + nld;
    __syncthreads();
    for (int s = 128; s > 0; s >>= 1) {
        if (tid < s) red[tid] += red[tid + s];
        __syncthreads();
    }
    if (tid == 0) partial[i] = red[0] + logf(sval[0]);
}

// ---------------- K3: final fixed-order reduction ----------------
__global__ void rnc_final(const float* __restrict__ partial, float* __restrict__ out) {
    __shared__ float red[256];
    int tid = threadIdx.x;
    float s = 0.0f;
    #pragma unroll
    for (int t = 0; t < 16; ++t) s += partial[tid + t * 256];
    red[tid] = s;
    __syncthreads();
    for (int k = 128; k > 0; k >>= 1) {
        if (tid < k) red[tid] += red[tid + k];
        __syncthreads();
    }
    if (tid == 0)
        out[0] = -red[0] / (float)((long long)N_TOT * (N_TOT - 1));
}

extern "C" void kernel_launch(void* const* d_in, const int* in_sizes, int n_in,
                              void* d_out, int out_size, void* d_ws, size_t ws_size,
                              hipStream_t stream) {
    (void)in_sizes; (void)n_in; (void)out_size; (void)ws_size;
    const float* z1     = (const float*)d_in[0];
    const float* z2     = (const float*)d_in[1];
    const float* labels = (const float*)d_in[2];

    char*  ws      = (char*)d_ws;
    float* lo      = (float*)ws;                                   // n*n f32 = 64 MB
    float* r       = (float*)(ws + (size_t)N_TOT * N_TOT * 4);
    float* llab    = r + N_TOT;
    float* partial = llab + N_TOT;

    rnc_prep<<<N_TOT / 256, 256, 0, stream>>>(z1, z2, labels, r, llab);
    // 65536 tiles, 8 waves (tiles) per 256-thread block
    rnc_gemm_logits<<<(256 * 256) / 8, 256, 0, stream>>>(z1, z2, r, lo);
    rnc_row<<<N_TOT, 256, 0, stream>>>(lo, llab, partial);
    rnc_final<<<1, 256, 0, stream>>>(partial, (float*)d_out);
}